// NMS_4_Pnetouts_67774583930889
// MI455X (gfx1250) — compile-verified
//
#include <hip/hip_runtime.h>
#include <cstdint>
#include <cstddef>

#define B_        8
#define N_        5000
#define H_        1024
#define W_        1024
#define MAX_OUT_  100
#define OUT_      24
#define IOU_THR_  0.7f
#define NEGV      (-1e30f)

#define NMS_BLOCK 512
#define CHUNK     10            // 512*10 = 5120 >= 5000
#define CROP_BLOCK 576          // 24*24 pixels

#define CROPS_ELEMS (B_*MAX_OUT_*OUT_*OUT_*3)   // 1,382,400

// ---- CDNA5 async global->LDS path (guarded; device pass only) ----
#if defined(__HIP_DEVICE_COMPILE__) && __has_builtin(__builtin_amdgcn_global_load_async_to_lds_b32)
#define HAVE_ASYNC_LDS 1
#else
#define HAVE_ASYNC_LDS 0
#endif

static __device__ __forceinline__ void wait_async_zero() {
#if defined(__HIP_DEVICE_COMPILE__)
#if __has_builtin(__builtin_amdgcn_s_wait_asynccnt)
  __builtin_amdgcn_s_wait_asynccnt(0);
#else
  asm volatile("s_wait_asynccnt 0" ::: "memory");
#endif
#endif
}

// =====================================================================
// Kernel 1: NMS. One block per batch. Boxes live in VGPRs; argmax via
// wave32 shfl_xor butterflies + one 16-entry LDS hop (2 barriers/round).
// =====================================================================
__global__ __launch_bounds__(NMS_BLOCK)
void nms_kernel(const float* __restrict__ rects, float* __restrict__ bb_out)
{
  const int b = blockIdx.x;
  const int t = threadIdx.x;
  const float* R = rects + (size_t)b * N_ * 5;

  float x1[CHUNK], y1[CHUNK], x2[CHUNK], y2[CHUNK], sc[CHUNK], ar[CHUNK];
  unsigned vmask = 0u;

#pragma unroll
  for (int k = 0; k < CHUNK; ++k) {
    int n = k * NMS_BLOCK + t;
    if (n < N_) {
      const float* r = R + (size_t)n * 5;
      x1[k] = r[0]; y1[k] = r[1]; x2[k] = r[2]; y2[k] = r[3]; sc[k] = r[4];
      ar[k] = (x2[k] - x1[k]) * (y2[k] - y1[k]);
      vmask |= (1u << k);
    } else {
      x1[k] = 0.f; y1[k] = 0.f; x2[k] = 0.f; y2[k] = 0.f;
      sc[k] = NEGV; ar[k] = 0.f;
    }
  }

  __shared__ float s_ps[NMS_BLOCK / 32];
  __shared__ int   s_pi[NMS_BLOCK / 32];
  __shared__ float s_resS;
  __shared__ int   s_resI;
  __shared__ int   s_sel[MAX_OUT_];
  __shared__ int   s_msk[MAX_OUT_];
  __shared__ int   s_fin[MAX_OUT_];

  const int lane = t & 31;
  const int wid  = t >> 5;

  // block-wide argmax with lowest-index tie-break; result in s_resS/s_resI
  auto block_argmax = [&](float bs, int bn) {
#pragma unroll
    for (int off = 16; off > 0; off >>= 1) {
      float so = __shfl_xor(bs, off);
      int   io = __shfl_xor(bn, off);
      if (so > bs || (so == bs && io < bn)) { bs = so; bn = io; }
    }
    if (lane == 0) { s_ps[wid] = bs; s_pi[wid] = bn; }
    __syncthreads();
    if (wid == 0) {
      float ps = (lane < NMS_BLOCK / 32) ? s_ps[lane] : NEGV;
      int   pi = (lane < NMS_BLOCK / 32) ? s_pi[lane] : 0x7fffffff;
#pragma unroll
      for (int off = 16; off > 0; off >>= 1) {
        float so = __shfl_xor(ps, off);
        int   io = __shfl_xor(pi, off);
        if (so > ps || (so == ps && io < pi)) { ps = so; pi = io; }
      }
      if (lane == 0) { s_resS = ps; s_resI = pi; }
    }
    __syncthreads();
  };

  // global (unmasked) argmax of raw scores -> fill index
  {
    float bs = NEGV; int bn = 0x7fffffff;
#pragma unroll
    for (int k = 0; k < CHUNK; ++k) {
      int n = k * NMS_BLOCK + t;
      float s = sc[k];                       // OOB slots hold NEGV
      if (s > bs || (s == bs && n < bn)) { bs = s; bn = n; }
    }
    block_argmax(bs, bn);
  }
  const int gmax = s_resI;                   // safe: next write is after a barrier

  for (int i = 0; i < MAX_OUT_; ++i) {
    float bs = NEGV; int bn = 0x7fffffff;
#pragma unroll
    for (int k = 0; k < CHUNK; ++k) {
      int n = k * NMS_BLOCK + t;
      float s = ((vmask >> k) & 1u) ? sc[k] : NEGV;
      if (s > bs || (s == bs && n < bn)) { bs = s; bn = n; }
    }
    block_argmax(bs, bn);
    const float sbest = s_resS;
    const int   jbest = s_resI;
    const bool  ok = (sbest > -1e29f);

    if (t == 0) { s_sel[i] = jbest; s_msk[i] = ok ? 1 : 0; }

    if (ok) {
      const float* rj = R + (size_t)jbest * 5;
      const float jx1 = rj[0], jy1 = rj[1], jx2 = rj[2], jy2 = rj[3];
      const float jar = (jx2 - jx1) * (jy2 - jy1);
#pragma unroll
      for (int k = 0; k < CHUNK; ++k) {
        float ix1 = fmaxf(jx1, x1[k]);
        float iy1 = fmaxf(jy1, y1[k]);
        float ix2 = fminf(jx2, x2[k]);
        float iy2 = fminf(jy2, y2[k]);
        float inter = fmaxf(ix2 - ix1, 0.f) * fmaxf(iy2 - iy1, 0.f);
        float iou = inter / (jar + ar[k] - inter);
        if (iou > IOU_THR_) vmask &= ~(1u << k);
      }
    }
  }

  __syncthreads();
  if (t < MAX_OUT_) s_fin[t] = s_msk[t] ? s_sel[t] : gmax;
  __syncthreads();
  if (t < MAX_OUT_ * 5) {
    int slot = t / 5;
    int c    = t - slot * 5;
    bb_out[(size_t)b * (MAX_OUT_ * 5) + t] = R[(size_t)s_fin[slot] * 5 + c];
  }
}

// =====================================================================
// Kernel 2: bilinear crop. One block per (batch,slot). The w*h*3 source
// region is staged into LDS via async global->LDS copies (ASYNCcnt),
// then each thread samples one output pixel (3 channels) from LDS.
// =====================================================================
__global__ __launch_bounds__(CROP_BLOCK)
void crop_kernel(const float* __restrict__ img, const float* __restrict__ bb,
                 float* __restrict__ crops)
{
  __shared__ float tile[48 * 48 * 3];        // 27,648 B

  const int blk = blockIdx.x;                // b*100 + slot
  const int b   = blk / MAX_OUT_;
  const int t   = threadIdx.x;

  const float* box = bb + (size_t)blk * 5;
  const int bx1 = (int)box[0];
  const int by1 = (int)box[1];
  const int bx2 = (int)box[2];
  const int by2 = (int)box[3];
  const int h = by2 - by1;
  const int w = bx2 - bx1;
  const int hh = min(max(h, 1), 48);         // defensive; h,w in [11,48] in practice
  const int ww = min(max(w, 1), 48);

  int r_base = min(max(by1 - 1, 0), H_ - 1);
  int c_base = min(max(bx1 - 1, 0), W_ - 1);
  if (r_base + hh > H_) r_base = H_ - hh;
  if (c_base + ww > W_) c_base = W_ - ww;

  const int rowlen = ww * 3;
  const int total  = hh * rowlen;
  const float* src = img + ((size_t)(b * H_ + r_base) * W_ + c_base) * 3;

  for (int e = t; e < total; e += CROP_BLOCK) {
    int r   = e / rowlen;
    int off = e - r * rowlen;
    const float* g = src + (size_t)r * (W_ * 3) + off;
#if HAVE_ASYNC_LDS
    typedef __attribute__((address_space(1))) int glb_int_t;
    typedef __attribute__((address_space(3))) int lds_int_t;
    __builtin_amdgcn_global_load_async_to_lds_b32(
        (glb_int_t*)(void*)g, (lds_int_t*)(void*)&tile[e], 0, 0);
#else
    tile[e] = *g;
#endif
  }
#if HAVE_ASYNC_LDS
  wait_async_zero();
#endif
  __syncthreads();

  const float sy = (float)h / (float)OUT_;
  const float sx = (float)w / (float)OUT_;

  if (t < OUT_ * OUT_) {
    const int iy = t / OUT_;
    const int ix = t - iy * OUT_;
    const float ryf = (float)iy * sy;
    const float rxf = (float)ix * sx;
    const int y0 = (int)floorf(ryf);
    const int x0 = (int)floorf(rxf);
    const float wy = ryf - (float)y0;
    const float wx = rxf - (float)x0;
    int y0c = min(max(y0, 0), hh - 1);
    int y1c = min(max(y0 + 1, 0), hh - 1);
    int x0c = min(max(x0, 0), ww - 1);
    int x1c = min(max(x0 + 1, 0), ww - 1);

    float* out = crops + (size_t)blk * (OUT_ * OUT_ * 3) + (size_t)t * 3;
#pragma unroll
    for (int c = 0; c < 3; ++c) {
      float A  = tile[y0c * rowlen + x0c * 3 + c];
      float Bq = tile[y0c * rowlen + x1c * 3 + c];
      float C  = tile[y1c * rowlen + x0c * 3 + c];
      float D  = tile[y1c * rowlen + x1c * 3 + c];
      float top = A * (1.0f - wx) + Bq * wx;
      float bot = C * (1.0f - wx) + D * wx;
      out[c] = top * (1.0f - wy) + bot * wy;
    }
  }
}

// =====================================================================
extern "C" void kernel_launch(void* const* d_in, const int* in_sizes, int n_in,
                              void* d_out, int out_size, void* d_ws, size_t ws_size,
                              hipStream_t stream)
{
  const float* rects = (const float*)d_in[0];  // (8,5000,5)
  const float* img   = (const float*)d_in[1];  // (8,1024,1024,3)
  float* out   = (float*)d_out;
  float* crops = out;                          // (8,100,24,24,3) flat
  float* bb    = out + CROPS_ELEMS;            // (8,100,5) flat

  nms_kernel<<<B_, NMS_BLOCK, 0, stream>>>(rects, bb);
  crop_kernel<<<B_ * MAX_OUT_, CROP_BLOCK, 0, stream>>>(img, bb, crops);
}